// LSTM_57887569216034
// MI455X (gfx1250) — compile-verified
//
#include <hip/hip_runtime.h>
#include <hip/hip_bf16.h>

#define T_STEPS 2048
#define BATCH   64
#define IN_DIM  32
#define HID     512
#define OUT_DIM 32
#define NWG     16      // one workgroup per hidden slice (persistent, one per WGP)
#define NTHREADS 256    // 8 wave32s
#define SLICE   32      // hidden units per WG
#define GCOLS   128     // gate columns per WG = 4 gates * SLICE

typedef float v2f __attribute__((ext_vector_type(2)));
typedef float v8f __attribute__((ext_vector_type(8)));

// LDS layout (floats):
//   Phh   : 512*128  = 65536   (W_hh slice, packed in WMMA-B lane order)
//   Pih   :  32*128  =  4096   (W_ih slice, same packing)
//   gates :  64*128  =  8192   (per-step gate GEMM result)
//   cst   :  64*32   =  2048   (persistent cell state slice)
#define SMEM_FLOATS (65536 + 4096 + 8192 + 2048)
#define SMEM_BYTES  (SMEM_FLOATS * 4)   // 319488 B <= 320 KB/WGP

__global__ void lstm_init_kernel(const float* __restrict__ h0,
                                 float* __restrict__ hbuf0,
                                 unsigned* __restrict__ ctr) {
    int i = blockIdx.x * blockDim.x + threadIdx.x;
    if (i == 0) *ctr = 0u;
    if (i < BATCH * HID) hbuf0[i] = h0[i];
}

__global__ void __launch_bounds__(NTHREADS, 1)
lstm_persistent_kernel(const float* __restrict__ inputs,
                       const float* __restrict__ c0,
                       const float* __restrict__ W_ih,
                       const float* __restrict__ W_hh,
                       const float* __restrict__ b_ih,
                       const float* __restrict__ b_hh,
                       const float* __restrict__ W_out,
                       const float* __restrict__ b_out,
                       float* __restrict__ out,
                       float* __restrict__ hbuf0,
                       float* __restrict__ hbuf1,
                       unsigned* __restrict__ ctr) {
    extern __shared__ float smem[];
    float* Phh   = smem;                    // 65536
    float* Pih   = Phh + 512 * GCOLS;       // 4096
    float* gates = Pih + IN_DIM * GCOLS;    // 8192
    float* cst   = gates + BATCH * GCOLS;   // 2048

    const int w    = blockIdx.x;            // slice id
    const int tid  = threadIdx.x;
    const int lane = tid & 31;              // wave32
    const int wave = tid >> 5;              // 0..7
    const int lm   = lane & 15;             // row-within-tile for A / col for B,C
    const int half = lane >> 4;             // K sub-pair selector for A/B operands
    const int n0   = wave * 16;             // this wave's gate-column offset
    const int ncol = n0 + lm;               // gate column 0..127

    // global gate-row for this lane's column: [i|f|g|o] blocks of 512
    const int grow = (ncol >> 5) * HID + w * SLICE + (ncol & 31);
    const float bias = b_ih[grow] + b_hh[grow];

    // --- pack W_hh slice into LDS in WMMA B-operand order ---------------
    // flat = ((k>>1)*128 + n)*2 + (k&1)  => lane reads float2 at
    //        ((2*kt + half)*128 + ncol)*2  == B[k..k+1][ncol]
    for (int idx = tid; idx < HID * GCOLS; idx += NTHREADS) {
        int j = idx & 1, rest = idx >> 1;
        int n = rest & (GCOLS - 1);
        int k = ((rest >> 7) << 1) | j;
        int gr = (n >> 5) * HID + w * SLICE + (n & 31);
        Phh[idx] = W_hh[gr * HID + k];
    }
    for (int idx = tid; idx < IN_DIM * GCOLS; idx += NTHREADS) {
        int j = idx & 1, rest = idx >> 1;
        int n = rest & (GCOLS - 1);
        int k = ((rest >> 7) << 1) | j;
        int gr = (n >> 5) * HID + w * SLICE + (n & 31);
        Pih[idx] = W_ih[gr * IN_DIM + k];
    }
    // persistent cell-state slice
    for (int e = tid; e < BATCH * SLICE; e += NTHREADS) {
        int b = e >> 5, u = e & 31;
        cst[e] = c0[b * HID + w * SLICE + u];
    }
    __syncthreads();

    for (int t = 0; t < T_STEPS; ++t) {
        const float* hcur  = (t & 1) ? hbuf1 : hbuf0;
        float*       hnext = (t & 1) ? hbuf0 : hbuf1;
        const float* xt    = inputs + (size_t)t * BATCH * IN_DIM;

        // C init = bias (per gate column, broadcast over batch rows)
        v8f acc[4];
#pragma unroll
        for (int m = 0; m < 4; ++m)
#pragma unroll
            for (int r = 0; r < 8; ++r) acc[m][r] = bias;

        // ---- x projection: K = 32 -> 8 k-tiles ----
#pragma unroll
        for (int kt = 0; kt < IN_DIM / 4; ++kt) {
            v2f bt = *(const v2f*)&Pih[((kt * 2 + half) * GCOLS + ncol) * 2];
#pragma unroll
            for (int m = 0; m < 4; ++m) {
                v2f at = *(const v2f*)&xt[(m * 16 + lm) * IN_DIM + kt * 4 + half * 2];
                acc[m] = __builtin_amdgcn_wmma_f32_16x16x4_f32(
                    false, at, false, bt, (short)0, acc[m], false, false);
            }
        }
        // ---- recurrence: K = 512 -> 128 k-tiles ----
#pragma unroll 4
        for (int kt = 0; kt < HID / 4; ++kt) {
            v2f bt = *(const v2f*)&Phh[((kt * 2 + half) * GCOLS + ncol) * 2];
#pragma unroll
            for (int m = 0; m < 4; ++m) {
                v2f at = *(const v2f*)&hcur[(m * 16 + lm) * HID + kt * 4 + half * 2];
                acc[m] = __builtin_amdgcn_wmma_f32_16x16x4_f32(
                    false, at, false, bt, (short)0, acc[m], false, false);
            }
        }

        // scatter gates to LDS (C layout: row = 16m + 8*half + r, col = ncol)
#pragma unroll
        for (int m = 0; m < 4; ++m)
#pragma unroll
            for (int r = 0; r < 8; ++r)
                gates[(m * 16 + half * 8 + r) * GCOLS + ncol] = acc[m][r];
        __syncthreads();

        // ---- elementwise LSTM cell: 64*32 elems / 256 threads ----
        for (int e = tid; e < BATCH * SLICE; e += NTHREADS) {
            int b = e >> 5, u = e & 31;
            float gi = gates[b * GCOLS + u];
            float gf = gates[b * GCOLS + 32 + u];
            float gg = gates[b * GCOLS + 64 + u];
            float go = gates[b * GCOLS + 96 + u];
            gi = 1.0f / (1.0f + __expf(-gi));
            gf = 1.0f / (1.0f + __expf(-gf));
            gg = tanhf(gg);
            go = 1.0f / (1.0f + __expf(-go));
            float cn = gf * cst[e] + gi * gg;
            cst[e] = cn;
            hnext[b * HID + w * SLICE + u] = go * tanhf(cn);
        }

        // ---- device-wide barrier (release h slice, acquire full h) ----
        __syncthreads();
        if (tid == 0) {
            __hip_atomic_fetch_add(ctr, 1u, __ATOMIC_RELEASE, __HIP_MEMORY_SCOPE_AGENT);
            const unsigned tgt = (unsigned)(NWG * (t + 1));
            while (__hip_atomic_load(ctr, __ATOMIC_ACQUIRE, __HIP_MEMORY_SCOPE_AGENT) < tgt)
                __builtin_amdgcn_s_sleep(2);
        }
        __syncthreads();

        // ---- fused output projection: this WG handles 4 batch rows ----
        {
            int bl = tid >> 6;            // 0..3
            int b  = w * 4 + bl;
            int o  = (tid >> 1) & 31;
            int hh = tid & 1;             // half of K, reduced via shuffle
            const float4* hv = (const float4*)(hnext + b * HID + hh * 256);
            const float4* wv = (const float4*)(W_out + o * HID + hh * 256);
            float s = 0.0f;
#pragma unroll 8
            for (int kk = 0; kk < 64; ++kk) {
                float4 a = hv[kk], c = wv[kk];
                s += a.x * c.x + a.y * c.y + a.z * c.z + a.w * c.w;
            }
            s += __shfl_xor(s, 1, 32);
            if (hh == 0)
                out[(size_t)t * BATCH * OUT_DIM + b * OUT_DIM + o] = s + b_out[o];
        }

        // prefetch next timestep's input slab into cache
        if (t + 1 < T_STEPS)
            __builtin_prefetch(inputs + (size_t)(t + 1) * BATCH * IN_DIM + tid * 8, 0, 1);
    }

    // ---- epilogue: h_T and c_T slices ----
    float* hT = out + (size_t)T_STEPS * BATCH * OUT_DIM;
    float* cT = hT + BATCH * HID;
    const float* hfin = (T_STEPS & 1) ? hbuf1 : hbuf0;   // T even -> hbuf0
    for (int e = tid; e < BATCH * SLICE; e += NTHREADS) {
        int b = e >> 5, u = e & 31;
        hT[b * HID + w * SLICE + u] = hfin[b * HID + w * SLICE + u];
        cT[b * HID + w * SLICE + u] = cst[e];
    }
}

extern "C" void kernel_launch(void* const* d_in, const int* in_sizes, int n_in,
                              void* d_out, int out_size, void* d_ws, size_t ws_size,
                              hipStream_t stream) {
    const float* inputs = (const float*)d_in[0];
    const float* h0     = (const float*)d_in[1];
    const float* c0     = (const float*)d_in[2];
    const float* W_ih   = (const float*)d_in[3];
    const float* W_hh   = (const float*)d_in[4];
    const float* b_ih   = (const float*)d_in[5];
    const float* b_hh   = (const float*)d_in[6];
    const float* W_out  = (const float*)d_in[7];
    const float* b_out  = (const float*)d_in[8];
    float* out = (float*)d_out;

    // workspace: [ctr pad 256B][hbuf0 64*512 f32][hbuf1 64*512 f32]
    unsigned* ctr = (unsigned*)d_ws;
    float* hbuf0 = (float*)((char*)d_ws + 256);
    float* hbuf1 = hbuf0 + BATCH * HID;

    lstm_init_kernel<<<dim3((BATCH * HID + 255) / 256), dim3(256), 0, stream>>>(
        h0, hbuf0, ctr);

    (void)hipFuncSetAttribute((const void*)lstm_persistent_kernel,
                              hipFuncAttributeMaxDynamicSharedMemorySize, SMEM_BYTES);

    lstm_persistent_kernel<<<dim3(NWG), dim3(NTHREADS), SMEM_BYTES, stream>>>(
        inputs, c0, W_ih, W_hh, b_ih, b_hh, W_out, b_out,
        out, hbuf0, hbuf1, ctr);
}